// RPNProposal_43800076485305
// MI455X (gfx1250) — compile-verified
//
#include <hip/hip_runtime.h>
#include <stdint.h>

#define N_ANCHORS 250000
#define PRE_NMS   6000
#define POST_NMS  300
#define IOU_THR   0.7f
#define NB        ((PRE_NMS + 31) / 32)   /* 188 words per bitmask row */
#define SORT_N    8192
#define ROWS_PER_BLK 256                  /* IoU: 256 rows x 32 cols per block */

// ---------- helpers ----------
__device__ __forceinline__ unsigned f2key(float f) {
    // order-preserving map: larger float -> larger unsigned
    unsigned b = __float_as_uint(f);
    return (b & 0x80000000u) ? ~b : (b | 0x80000000u);
}
// -inf maps to 0x007FFFFF; any finite score maps >= 0x00800000
#define VALID_KEY_MIN 0x00800000u

__global__ void k_zero(unsigned* p, int n) {
    int i = blockIdx.x * blockDim.x + threadIdx.x;
    if (i < n) p[i] = 0u;
}

// ---------- 1. decode + clip + validity + sortable key ----------
__global__ void k_decode(const float4* __restrict__ deltas,
                         const float4* __restrict__ anchors,
                         const float* __restrict__ scores,
                         const int* __restrict__ imh, const int* __restrict__ imw,
                         const int* __restrict__ stridep,
                         float4* __restrict__ boxes, unsigned* __restrict__ keys) {
    int i = blockIdx.x * blockDim.x + threadIdx.x;
    if (i >= N_ANCHORS) return;
    // gfx1250 prefetch path (lowers to global_prefetch_b8)
    __builtin_prefetch(&anchors[i + 2048 < N_ANCHORS ? i + 2048 : i], 0, 0);
    __builtin_prefetch(&deltas[i + 2048 < N_ANCHORS ? i + 2048 : i], 0, 0);

    float H = (float)imh[0], W = (float)imw[0], S = (float)stridep[0];
    float4 a = anchors[i];   // ymin xmin ymax xmax
    float4 d = deltas[i];    // ty tx th tw
    float h = a.z - a.x, w = a.w - a.y;
    float cy = a.x + 0.5f * h, cx = a.y + 0.5f * w;
    float ncy = d.x * h + cy, ncx = d.y * w + cx;
    float nh = expf(d.z) * h, nw = expf(d.w) * w;
    float ymin = fminf(fmaxf(ncy - 0.5f * nh, 0.f), H);
    float xmin = fminf(fmaxf(ncx - 0.5f * nw, 0.f), W);
    float ymax = fminf(fmaxf(ncy + 0.5f * nh, 0.f), H);
    float xmax = fminf(fmaxf(ncx + 0.5f * nw, 0.f), W);
    bool valid = (ymax - ymin >= S) && (xmax - xmin >= S);
    float ms = valid ? scores[i] : -__builtin_inff();
    boxes[i] = make_float4(ymin, xmin, ymax, xmax);
    keys[i] = f2key(ms);
}

// ---------- 2. exact radix-select (two 16-bit passes) ----------
// st[0]=h* st[1]=G1 st[2]=T st[3]=cntGT st[4]=need_eq st[5]=ctrGT st[6]=ctrEQ
__global__ void k_hist_hi(const unsigned* __restrict__ keys, unsigned* __restrict__ hist) {
    int i = blockIdx.x * blockDim.x + threadIdx.x;
    if (i < N_ANCHORS) atomicAdd(&hist[keys[i] >> 16], 1u);
}
__global__ void k_hist_lo(const unsigned* __restrict__ keys, const unsigned* __restrict__ st,
                          unsigned* __restrict__ hist) {
    int i = blockIdx.x * blockDim.x + threadIdx.x;
    if (i < N_ANCHORS) {
        unsigned k = keys[i];
        if ((k >> 16) == st[0]) atomicAdd(&hist[k & 0xFFFFu], 1u);
    }
}
__global__ void k_scan(const unsigned* __restrict__ hist, unsigned* st, int pass) {
    __shared__ unsigned cs[1024];
    int t = threadIdx.x;
    unsigned target = (unsigned)PRE_NMS;
    if (pass == 2) target = (unsigned)PRE_NMS - st[1];
    unsigned s = 0;
    int base = t * 64;
    for (int b = 0; b < 64; ++b) s += hist[base + b];
    cs[t] = s;
    __syncthreads();
    if (t == 0) {
        unsigned acc = 0; int cstar = 0;
        for (int c = 1023; c >= 0; --c) { acc += cs[c]; if (acc >= target) { cstar = c; break; } }
        unsigned above = acc - cs[cstar];   // strictly above chunk c*
        unsigned acc2 = above; int bstar = cstar * 64;
        for (int b = cstar * 64 + 63; b >= cstar * 64; --b) {
            unsigned h = hist[b];
            acc2 += h;
            if (acc2 >= target) { bstar = b; above = acc2 - h; break; }
        }
        if (pass == 1) { st[0] = (unsigned)bstar; st[1] = above; }
        else {
            st[2] = (st[0] << 16) | (unsigned)bstar;      // exact 32-bit threshold T
            st[3] = st[1] + above;                        // count strictly > T
            st[4] = (unsigned)PRE_NMS - st[3];            // #equal-to-T to admit
            st[5] = 0; st[6] = 0;                         // gather counters
        }
    }
}
__global__ void k_gather(const unsigned* __restrict__ keys, unsigned* st,
                         unsigned long long* __restrict__ sel) {
    int i = blockIdx.x * blockDim.x + threadIdx.x;
    if (i >= N_ANCHORS) return;
    unsigned T = st[2], cntGT = st[3], need = st[4];
    unsigned k = keys[i];
    unsigned long long comp = ((unsigned long long)k << 32) | (unsigned)~(unsigned)i;
    if (k > T) {
        unsigned p = atomicAdd(&st[5], 1u);
        sel[p] = comp;
    } else if (k == T) {
        unsigned p = atomicAdd(&st[6], 1u);
        if (p < need) sel[cntGT + p] = comp;
    }
}

// ---------- 3. bitonic sort of 8192 x u64 in LDS (descending, stable by idx) ----------
__global__ __launch_bounds__(1024)
void k_sort(const unsigned long long* __restrict__ sel,
            unsigned* __restrict__ sorted_idx, unsigned* __restrict__ valid_bits) {
    __shared__ unsigned long long a[SORT_N];   // 64KB, fits 320KB/WGP LDS
    int tid = threadIdx.x;
    for (int e = tid; e < SORT_N; e += 1024) a[e] = (e < PRE_NMS) ? sel[e] : 0ULL;
    __syncthreads();
    for (int k = 2; k <= SORT_N; k <<= 1)
        for (int j = k >> 1; j > 0; j >>= 1) {
            for (int i = tid; i < SORT_N; i += 1024) {
                int l = i ^ j;
                if (l > i) {
                    bool desc = ((i & k) == 0);
                    unsigned long long x = a[i], y = a[l];
                    if ((x < y) == desc) { a[i] = y; a[l] = x; }
                }
            }
            __syncthreads();
        }
    for (int e = tid; e < PRE_NMS; e += 1024)
        sorted_idx[e] = ~(unsigned)(a[e] & 0xFFFFFFFFull);
    for (int w = tid; w < NB; w += 1024) {
        unsigned bits = 0;
        for (int b = 0; b < 32; ++b) {
            int e = w * 32 + b;
            if (e < PRE_NMS && (unsigned)(a[e] >> 32) >= VALID_KEY_MIN) bits |= 1u << b;
        }
        valid_bits[w] = bits;
    }
}

__global__ void k_gather_boxes(const float4* __restrict__ boxes,
                               const unsigned* __restrict__ sorted_idx,
                               float4* __restrict__ sboxes) {
    int r = blockIdx.x * blockDim.x + threadIdx.x;
    if (r < PRE_NMS) sboxes[r] = boxes[sorted_idx[r]];
}

// ---------- 4. parallel IoU bitmask: 256 rows x 32 cols per block ----------
// Column tile staged once in LDS via gfx1250 async global->LDS copy (ASYNCcnt),
// then reused by all 8 waves of the block.
__global__ __launch_bounds__(ROWS_PER_BLK)
void k_iou(const float4* __restrict__ sboxes, unsigned* __restrict__ mask) {
    int BI = blockIdx.x;                 // row stripe: rows [BI*256, BI*256+256)
    int bj = blockIdx.y;                 // col tile:  cols [bj*32,  bj*32+32)
    int j0 = bj * 32;
    int i0 = BI * ROWS_PER_BLK;
    if (j0 + 31 <= i0) return;           // entire tile below diagonal (pre-zeroed)

    int t = threadIdx.x;
    __shared__ float4 cols[32];
    if (t < 32) {
        int jc = j0 + t;
        if (jc < PRE_NMS) {
            unsigned lds_off = (unsigned)(uintptr_t)&cols[t];  // low 32 bits = LDS byte offset
            unsigned long long ga = (unsigned long long)(uintptr_t)(sboxes + jc);
            asm volatile("global_load_async_to_lds_b128 %0, %1, off"
                         :: "v"(lds_off), "v"(ga) : "memory");
        }
        asm volatile("s_wait_asynccnt 0" ::: "memory");
    }
    __syncthreads();

    int i = i0 + t;
    if (i >= PRE_NMS) return;
    float4 bx = sboxes[i];
    float areai = (bx.z - bx.x) * (bx.w - bx.y);
    unsigned bits = 0;
    #pragma unroll 4
    for (int c = 0; c < 32; ++c) {
        int j = j0 + c;
        if (j < PRE_NMS && j > i) {
            float4 by = cols[c];
            float areaj = (by.z - by.x) * (by.w - by.y);
            float iy = fmaxf(0.f, fminf(bx.z, by.z) - fmaxf(bx.x, by.x));
            float ix = fmaxf(0.f, fminf(bx.w, by.w) - fmaxf(bx.y, by.y));
            float inter = iy * ix;
            float iou = inter / (areai + areaj - inter + 1e-9f);
            if (iou > IOU_THR) bits |= 1u << c;
        }
    }
    mask[(unsigned)i * NB + bj] = bits;
}

// ---------- 5. sequential NMS scan (single wave32) + compaction ----------
__global__ __launch_bounds__(32)
void k_nms(const unsigned* __restrict__ mask, const unsigned* __restrict__ valid_bits,
           const float4* __restrict__ sboxes, float4* __restrict__ out) {
    __shared__ unsigned rem[NB];
    __shared__ unsigned vb[NB];
    int t = threadIdx.x;
    for (int w = t; w < NB; w += 32) { rem[w] = 0u; vb[w] = valid_bits[w]; }
    __syncthreads();
    for (int i = 0; i < PRE_NMS; ++i) {
        // hide next row's latency behind this iteration (global_prefetch_b8)
        if (i + 1 < PRE_NMS)
            __builtin_prefetch(&mask[(unsigned)(i + 1) * NB + t * 6], 0, 0);
        unsigned word = rem[i >> 5];
        bool alive = (((vb[i >> 5] >> (i & 31)) & 1u) != 0u) &&
                     (((word >> (i & 31)) & 1u) == 0u);
        if (alive) {
            const unsigned* row = &mask[(unsigned)i * NB];
            for (int w = t; w < NB; w += 32) rem[w] |= row[w];
        }
        __syncthreads();
    }
    if (t == 0) {
        int rank = 0;
        for (int i = 0; i < PRE_NMS && rank < POST_NMS; ++i) {
            bool keep = (((vb[i >> 5] >> (i & 31)) & 1u) != 0u) &&
                        (((rem[i >> 5] >> (i & 31)) & 1u) == 0u);
            if (keep) out[rank++] = sboxes[i];
        }
    }
}

// ---------- launcher ----------
extern "C" void kernel_launch(void* const* d_in, const int* in_sizes, int n_in,
                              void* d_out, int out_size, void* d_ws, size_t ws_size,
                              hipStream_t stream) {
    (void)in_sizes; (void)n_in; (void)out_size; (void)ws_size;
    const float4* deltas  = (const float4*)d_in[0];
    const float4* anchors = (const float4*)d_in[1];
    const float*  scores  = (const float*)d_in[2];
    const int* imh = (const int*)d_in[3];
    const int* imw = (const int*)d_in[4];
    const int* str = (const int*)d_in[5];

    // workspace carve-out (~10 MB total)
    char* w = (char*)d_ws;
    size_t o = 0;
    auto carve = [&](size_t bytes) { o = (o + 255) & ~(size_t)255; size_t r = o; o += bytes; return r; };
    float4*             boxes      = (float4*)(w + carve((size_t)N_ANCHORS * 16));
    unsigned*           keys       = (unsigned*)(w + carve((size_t)N_ANCHORS * 4));
    unsigned*           hist       = (unsigned*)(w + carve(65536 * 4));
    unsigned*           st         = (unsigned*)(w + carve(64 * 4));
    unsigned long long* sel        = (unsigned long long*)(w + carve((size_t)SORT_N * 8));
    unsigned*           sorted_idx = (unsigned*)(w + carve((size_t)PRE_NMS * 4));
    unsigned*           valid_bits = (unsigned*)(w + carve((size_t)NB * 4));
    float4*             sboxes     = (float4*)(w + carve((size_t)PRE_NMS * 16));
    unsigned*           maskbuf    = (unsigned*)(w + carve((size_t)PRE_NMS * NB * 4));

    const int TB = 256;
    int gN = (N_ANCHORS + TB - 1) / TB;

    // zero output (harness poisons it) and mask
    k_zero<<<(POST_NMS * 4 + TB - 1) / TB, TB, 0, stream>>>((unsigned*)d_out, POST_NMS * 4);
    k_zero<<<((int)(PRE_NMS * NB) + TB - 1) / TB, TB, 0, stream>>>(maskbuf, PRE_NMS * NB);

    // decode + keys
    k_decode<<<gN, TB, 0, stream>>>(deltas, anchors, scores, imh, imw, str, boxes, keys);

    // radix select: pass 1 (hi 16 bits)
    k_zero<<<(65536 + TB - 1) / TB, TB, 0, stream>>>(hist, 65536);
    k_hist_hi<<<gN, TB, 0, stream>>>(keys, hist);
    k_scan<<<1, 1024, 0, stream>>>(hist, st, 1);
    // pass 2 (lo 16 bits within winning hi bin)
    k_zero<<<(65536 + TB - 1) / TB, TB, 0, stream>>>(hist, 65536);
    k_hist_lo<<<gN, TB, 0, stream>>>(keys, st, hist);
    k_scan<<<1, 1024, 0, stream>>>(hist, st, 2);

    // gather the PRE_NMS winners, sort them descending (stable by index)
    k_gather<<<gN, TB, 0, stream>>>(keys, st, sel);
    k_sort<<<1, 1024, 0, stream>>>(sel, sorted_idx, valid_bits);
    k_gather_boxes<<<(PRE_NMS + TB - 1) / TB, TB, 0, stream>>>(boxes, sorted_idx, sboxes);

    // IoU bitmask: 256 rows x 32 cols per block, col tile async-staged in LDS
    dim3 giou((PRE_NMS + ROWS_PER_BLK - 1) / ROWS_PER_BLK, NB);
    k_iou<<<giou, ROWS_PER_BLK, 0, stream>>>(sboxes, maskbuf);
    k_nms<<<1, 32, 0, stream>>>(maskbuf, valid_bits, sboxes, (float4*)d_out);
}